// diRNN_13365938225207
// MI455X (gfx1250) — compile-verified
//
#include <hip/hip_runtime.h>
#include <hip/hip_bf16.h>

typedef __attribute__((ext_vector_type(2))) float v2f;
typedef __attribute__((ext_vector_type(8))) float v8f;

constexpr int NX = 512, NU = 32, NY = 8, LL = 2, BB = 64, TT = 512;
constexpr int NCG = 16;            // column groups (WGPs per batch group)
constexpr int NBG = 4;             // batch groups
constexpr int COLS = NX / NCG;     // 32 columns per WGP
constexpr int ROWS = BB / NBG;     // 16 batch rows per group

// ---------------- f32 WMMA helper: D = A(16x4) * B(4x16) + C ----------------
__device__ inline v8f vzero8() {
  v8f z = {0.f,0.f,0.f,0.f,0.f,0.f,0.f,0.f};
  return z;
}
__device__ inline v8f wmma4(v2f a, v2f b, v8f c) {
  return __builtin_amdgcn_wmma_f32_16x16x4_f32(false, a, false, b, (short)0, c,
                                               false, false);
}

// ---------------- inter-WGP barrier (sense-reversing, agent scope) ----------
__device__ inline void group_barrier(int* cnt, int* gen, int members) {
  __threadfence();           // release: make this block's global stores visible
  __syncthreads();
  if (threadIdx.x == 0) {
    int phase = __hip_atomic_load(gen, __ATOMIC_RELAXED, __HIP_MEMORY_SCOPE_AGENT);
    int prev  = __hip_atomic_fetch_add(cnt, 1, __ATOMIC_ACQ_REL, __HIP_MEMORY_SCOPE_AGENT);
    if (prev == members - 1) {
      __hip_atomic_store(cnt, 0, __ATOMIC_RELAXED, __HIP_MEMORY_SCOPE_AGENT);
      __hip_atomic_fetch_add(gen, 1, __ATOMIC_RELEASE, __HIP_MEMORY_SCOPE_AGENT);
    } else {
      while (__hip_atomic_load(gen, __ATOMIC_ACQUIRE, __HIP_MEMORY_SCOPE_AGENT) == phase) {
        __builtin_amdgcn_s_sleep(1);
      }
    }
  }
  __syncthreads();
  __threadfence();           // acquire: invalidate stale lines before reads
}

// ---------------- Newton-Schulz inversion helpers ---------------------------
__global__ void init_x_kernel(const float* __restrict__ E, float* __restrict__ X) {
  int idx = blockIdx.x * 256 + threadIdx.x;          // 2*512*512
  if (idx >= LL * NX * NX) return;
  int m = idx % (NX * NX);
  float v = -E[idx];
  if ((m / NX) == (m % NX)) v += 2.0f;               // X0 = 2I - E
  X[idx] = v;
}

// mode 0: C = A@B     mode 1: C = 2I - A@B      (per l in blockIdx.z)
__global__ void ns_gemm_kernel(const float* __restrict__ A, const float* __restrict__ B,
                               float* __restrict__ C, int mode) {
  __shared__ float As[16][16];
  __shared__ float Bs[16][17];
  const int l = blockIdx.z;
  const float* Ab = A + (size_t)l * NX * NX;
  const float* Bb = B + (size_t)l * NX * NX;
  float* Cb = C + (size_t)l * NX * NX;
  const int ty = threadIdx.y, tx = threadIdx.x;
  const int row = blockIdx.y * 16 + ty, col = blockIdx.x * 16 + tx;
  float acc = 0.f;
  for (int kb = 0; kb < NX; kb += 16) {
    As[ty][tx] = Ab[(size_t)row * NX + kb + tx];
    Bs[ty][tx] = Bb[(size_t)(kb + ty) * NX + col];
    __syncthreads();
#pragma unroll
    for (int k = 0; k < 16; ++k) acc += As[ty][k] * Bs[k][tx];
    __syncthreads();
  }
  if (mode) { acc = -acc; if (row == col) acc += 2.0f; }
  Cb[(size_t)row * NX + col] = acc;
}

// ---------------- misc prep -------------------------------------------------
__global__ void transpose_u_kernel(const float* __restrict__ u, float* __restrict__ uT) {
  int idx = blockIdx.x * 256 + threadIdx.x;          // B*T*NU = 1M
  if (idx >= BB * TT * NU) return;
  int b = idx / (TT * NU);
  int r = idx % (TT * NU);
  int t = r / NU, k = r % NU;
  uT[idx] = u[((size_t)b * NU + k) * TT + t];        // (B,NU,T) -> (B,T,NU)
}

__global__ void init_misc_kernel(float* __restrict__ out, const float* __restrict__ Cb,
                                 float* __restrict__ hbuf, int* __restrict__ bar) {
  int idx = blockIdx.x * 256 + threadIdx.x;          // 262144 threads
  if (idx < BB * NY * TT) {
    int y = (idx / TT) % NY;
    out[idx] = Cb[y];                                // covers t = T-1 (sel==0) too
  }
  if (idx < BB * NX) hbuf[idx] = 0.0f;               // h0 = 0
  if (idx < 256) bar[idx] = 0;                       // reset barriers every launch
}

// ---------------- persistent scan kernel ------------------------------------
// grid = 64 blocks (4 batch groups x 16 column groups), 256 threads (8 waves)
__global__ __launch_bounds__(256) void scan_kernel(
    const float* __restrict__ Hw, const float* __restrict__ Hb,
    const float* __restrict__ Kw, const float* __restrict__ Cw,
    const float* __restrict__ Cb, const float* __restrict__ Einv,
    const float* __restrict__ uT, float* __restrict__ hbuf,
    float* __restrict__ ehbuf, int* __restrict__ barmem,
    float* __restrict__ out) {
  extern __shared__ float smem[];
  float* HwTs  = smem;               // [2][2][256][16][2] = 32768 f (Hw^T slices)
  float* EinvS = HwTs + 32768;       // 32768 f (Einv column slices)
  float* KwTs  = EinvS + 32768;      // 2048 f  (Kw^T slices)
  float* Cws   = KwTs + 2048;        // 4096 f  (full Cw)
  float* Hbs   = Cws + 4096;         // 64 f
  float* Cbs   = Hbs + 64;           // 8 f
  float* red   = Cbs + 8;            // 2048 f  (partial-C reduction)
  // total 73800 floats = 295200 B  (< 320KB WGP LDS)

  const int tid  = threadIdx.x;
  const int cg   = blockIdx.x % NCG;
  const int bg   = blockIdx.x / NCG;
  const int col0 = cg * COLS;
  const int row0 = bg * ROWS;

  // ---- one-time: stage weight slices into LDS, swizzled as WMMA B-fragments.
  // B frag (4x16 f32): lane = n(0..15) + 16*khalf; v0,v1 = B[k+2*khalf+{0,1}][n]
  // LDS layout per 16-col tile: ((k>>1)*16 + n)*2 + (k&1)
  for (int i = tid; i < 2 * 16384; i += 256) {       // Hw^T: B[k][n]=Hw[l][col0+n][k]
    int l = i >> 14, r = i & 16383;
    int n = r >> 9, k = r & 511;
    float v = Hw[(size_t)l * NX * NX + (size_t)(col0 + n) * NX + k];
    HwTs[l * 16384 + (n >> 4) * 8192 + (k >> 1) * 32 + (n & 15) * 2 + (k & 1)] = v;
  }
  for (int i = tid; i < 2 * 16384; i += 256) {       // Einv slice for layer l: Einv[(l+1)%2][k][col0+n]
    int l = i >> 14, r = i & 16383;
    int k = r >> 5, n = r & 31;
    float v = Einv[(size_t)((l + 1) & 1) * NX * NX + (size_t)k * NX + (col0 + n)];
    EinvS[l * 16384 + (n >> 4) * 8192 + (k >> 1) * 32 + (n & 15) * 2 + (k & 1)] = v;
  }
  for (int i = tid; i < 2 * 1024; i += 256) {        // Kw^T: B[k][n]=Kw[l][col0+n][k]
    int l = i >> 10, r = i & 1023;
    int n = r >> 5, k = r & 31;
    float v = Kw[(size_t)l * NX * NU + (size_t)(col0 + n) * NU + k];
    KwTs[l * 1024 + (n >> 4) * 512 + (k >> 1) * 32 + (n & 15) * 2 + (k & 1)] = v;
  }
  for (int i = tid; i < NY * NX; i += 256) Cws[i] = Cw[i];
  if (tid < 2 * COLS) Hbs[tid] = Hb[(tid >> 5) * NX + col0 + (tid & 31)];
  if (tid < NY) Cbs[tid] = Cb[tid];
  __syncthreads();

  const int wave  = tid >> 5, lane = tid & 31;
  const int tile  = wave >> 2;         // which 16-col tile (0,1)
  const int kw    = wave & 3;          // 4-way K split
  const int khalf = lane >> 4;         // A frag: v0,v1 = A[m][k+2*khalf+{0,1}]
  const int bn    = lane & 15;
  const int arow  = row0 + (lane & 15);
  int* bar_cnt = barmem + bg * 64;
  int* bar_gen = bar_cnt + 16;

  for (int t = 0; t < TT - 1; ++t) {
    for (int l = 0; l < LL; ++l) {
      // ============ GEMM1: x = h @ Hw[l]^T (+ u_t @ Kw[l]^T + Hb[l]); relu ==
      v8f c = vzero8();
      if (kw == 0) {                                 // fold input GEMM (K=32)
        const float* ub = uT + (size_t)arow * (TT * NU) + (size_t)t * NU + 2 * khalf;
        const float* kb = KwTs + l * 1024 + tile * 512 + khalf * 32 + bn * 2;
#pragma unroll
        for (int k = 0; k < NU; k += 4) {
          v2f a = *(const v2f*)(ub + k);
          v2f b = *(const v2f*)(kb + (k >> 1) * 32);
          c = wmma4(a, b, c);
        }
      }
      {
        const float* ab = hbuf + (size_t)arow * NX + 2 * khalf + kw * 128;
        const float* bb = HwTs + l * 16384 + tile * 8192 + kw * 2048 + khalf * 32 + bn * 2;
#pragma unroll 8
        for (int k = 0; k < 128; k += 4) {
          v2f a = *(const v2f*)(ab + k);
          v2f b = *(const v2f*)(bb + (k >> 1) * 32);
          c = wmma4(a, b, c);
        }
      }
      *(v8f*)(red + tid * 8) = c;
      __syncthreads();
      if (kw == 0) {                                 // reduce 4 K-partials + epilogue
        v8f s = *(const v8f*)(red + ((tile * 4 + 0) * 32 + lane) * 8);
        s    += *(const v8f*)(red + ((tile * 4 + 1) * 32 + lane) * 8);
        s    += *(const v8f*)(red + ((tile * 4 + 2) * 32 + lane) * 8);
        s    += *(const v8f*)(red + ((tile * 4 + 3) * 32 + lane) * 8);
        float bias = Hbs[l * COLS + tile * 16 + bn];
        union { v8f v; float f[8]; } su; su.v = s;
        float* eb = ehbuf + (size_t)col0 + tile * 16 + bn;
#pragma unroll
        for (int r = 0; r < 8; ++r) {                // C frag: row = r+8*khalf, col = bn
          float v = fmaxf(su.f[r] + bias, 0.f);
          eb[(size_t)(row0 + r + 8 * khalf) * NX] = v;
        }
      }
      group_barrier(bar_cnt, bar_gen, NCG);
      // ============ GEMM2: h' = eh @ Einv[(l+1)%2] ==========================
      c = vzero8();
      {
        const float* ab = ehbuf + (size_t)arow * NX + 2 * khalf + kw * 128;
        const float* bb = EinvS + l * 16384 + tile * 8192 + kw * 2048 + khalf * 32 + bn * 2;
#pragma unroll 8
        for (int k = 0; k < 128; k += 4) {
          v2f a = *(const v2f*)(ab + k);
          v2f b = *(const v2f*)(bb + (k >> 1) * 32);
          c = wmma4(a, b, c);
        }
      }
      *(v8f*)(red + tid * 8) = c;
      __syncthreads();
      if (kw == 0) {
        v8f s = *(const v8f*)(red + ((tile * 4 + 0) * 32 + lane) * 8);
        s    += *(const v8f*)(red + ((tile * 4 + 1) * 32 + lane) * 8);
        s    += *(const v8f*)(red + ((tile * 4 + 2) * 32 + lane) * 8);
        s    += *(const v8f*)(red + ((tile * 4 + 3) * 32 + lane) * 8);
        union { v8f v; float f[8]; } su; su.v = s;
        float* hb2 = hbuf + (size_t)col0 + tile * 16 + bn;
#pragma unroll
        for (int r = 0; r < 8; ++r)
          hb2[(size_t)(row0 + r + 8 * khalf) * NX] = su.f[r];
      }
      group_barrier(bar_cnt, bar_gen, NCG);
      // ============ projection of layer-0 state (sel[t]) ====================
      if (l == 0) {                                  // block cg handles batch row row0+cg
        const float* hr  = hbuf + (size_t)(row0 + cg) * NX;
        const float* cwr = Cws + wave * NX;          // wave == y (NY==8 waves)
        float sacc = 0.f;
#pragma unroll 4
        for (int i2 = 0; i2 < 16; ++i2)
          sacc += hr[lane + 32 * i2] * cwr[lane + 32 * i2];
#pragma unroll
        for (int off = 16; off > 0; off >>= 1) sacc += __shfl_down(sacc, off, 32);
        if (lane == 0)
          out[((size_t)(row0 + cg) * NY + wave) * TT + t] = sacc + Cbs[wave];
      }
    }
  }
}

// ---------------- host side -------------------------------------------------
extern "C" void kernel_launch(void* const* d_in, const int* in_sizes, int n_in,
                              void* d_out, int out_size, void* d_ws, size_t ws_size,
                              hipStream_t stream) {
  const float* u  = (const float*)d_in[0];
  const float* E  = (const float*)d_in[1];
  const float* Hw = (const float*)d_in[2];
  const float* Hb = (const float*)d_in[3];
  const float* Kw = (const float*)d_in[4];
  const float* Cw = (const float*)d_in[5];
  const float* Cb = (const float*)d_in[6];
  float* out = (float*)d_out;
  float* W   = (float*)d_ws;

  float* Xa  = W;                    // 2*512*512  (Newton-Schulz ping)
  float* Xb  = W + 524288;           // pong
  float* Cm  = W + 1048576;          // NS temp
  float* uT  = W + 1572864;          // B*T*NU
  float* hb  = W + 2621440;          // 64x512 hidden state
  float* eh  = W + 2654208;          // 64x512 relu activations
  int*   bar = (int*)(W + 2686976);  // barrier counters (4 groups x 64 ints)

  // Einv via Newton-Schulz: X0 = 2I - E; X <- X(2I - E X), 7 quadratic steps
  init_x_kernel<<<2048, 256, 0, stream>>>(E, Xa);
  for (int it = 0; it < 7; ++it) {
    ns_gemm_kernel<<<dim3(32, 32, 2), dim3(16, 16), 0, stream>>>(E, Xa, Cm, 1);
    ns_gemm_kernel<<<dim3(32, 32, 2), dim3(16, 16), 0, stream>>>(Xa, Cm, Xb, 0);
    float* tmp = Xa; Xa = Xb; Xb = tmp;
  }

  transpose_u_kernel<<<4096, 256, 0, stream>>>(u, uT);
  init_misc_kernel<<<1024, 256, 0, stream>>>(out, Cb, hb, bar);

  const int smem_bytes = 73800 * 4;  // 295200 B of the 320KB WGP LDS
  hipFuncSetAttribute((const void*)scan_kernel,
                      hipFuncAttributeMaxDynamicSharedMemorySize, smem_bytes);
  scan_kernel<<<NBG * NCG, 256, smem_bytes, stream>>>(Hw, Hb, Kw, Cw, Cb, Xa, uT,
                                                      hb, eh, bar, out);
}